// ComparisonLoss_17987323036122
// MI455X (gfx1250) — compile-verified
//
#include <hip/hip_runtime.h>

// ComparisonLoss (GHM-balanced BCE) for MI455X / gfx1250, wave32.
// pass1: column sums (Σw, Σt·w, Σt) via V_WMMA_F32_16X16X4_F32 with B=ones.
// pass2: per-class scalars. pass3: elementwise bce*w (L2-resident re-read) -> scalar.
// Bin tests on g=|sigmoid(p)-t| are done in logit space: g>=0.9 <=> z>=ln9,
// g<0.1 <=> z<-ln9 with z = (t?-p:p), removing the f32 divide/exp chain.

typedef __attribute__((ext_vector_type(2))) float v2f;
typedef __attribute__((ext_vector_type(8))) float v8f;

#define NB 262144
#define NC 40
#define NBf 262144.0f
#define LN9 2.1972246f   // logit(0.9); sigma(z)>=0.9 <=> z>=LN9, sigma(z)<0.1 <=> z<-LN9

// ---- pass 1 geometry: 3 class-groups of 16, 1024 waves per group ----
#define P1_WAVES_PER_GROUP 1024
#define P1_ROWS_PER_WAVE (NB / P1_WAVES_PER_GROUP)   // 256
#define P1_BLOCK 256                                  // 8 waves/block
#define P1_GRID ((3 * P1_WAVES_PER_GROUP) / (P1_BLOCK / 32))  // 384 blocks

// ---- pass 3 geometry: stride multiple of 40 so class is loop-invariant ----
#define P3_BLOCK 320
#define P3_GRID 1024   // 327680 threads; 10485760/327680 = 32 iters

// ws layout (floats): [0:40) sumW  [40:80) sumTW  [80:120) sumT
//                     [120) pos_gt [160) neg_gt [200) smaj [240) smin [280) minflag

__global__ void ghm_init(float* __restrict__ ws, float* __restrict__ out) {
    int i = threadIdx.x;
    if (i < 3 * NC) ws[i] = 0.0f;
    if (i == 0) out[0] = 0.0f;
}

__global__ __launch_bounds__(P1_BLOCK)
void ghm_pass1(const float* __restrict__ pred, const float* __restrict__ target,
               const float* __restrict__ rand_mat, const float* __restrict__ drate,
               float* __restrict__ ws) {
    const int tid  = threadIdx.x;
    const int lane = tid & 31;
    const int wid  = blockIdx.x * (P1_BLOCK / 32) + (tid >> 5);   // 0..3071
    const int g     = wid % 3;          // class group
    const int chunk = wid / 3;          // batch chunk
    const int clsLocal = lane & 15;     // A-matrix row M = class within group
    const int half     = lane >> 4;     // 0: K={0,1}  1: K={2,3}
    const int c = g * 16 + clsLocal;
    const bool valid = (c < NC);
    const float dr = valid ? drate[c] : 0.0f;
    const int rowbase = chunk * P1_ROWS_PER_WAVE;

    v8f accW  = {};   // Σ w        per class (rows of D)
    v8f accTW = {};   // Σ t*w
    v8f accT  = {};   // Σ t
    const v2f bones = {1.0f, 1.0f};   // B = ones => D[m][n] = Σ_k A[m][k]

    for (int r = rowbase; r < rowbase + P1_ROWS_PER_WAVE; r += 4) {
        const int row0 = r + half * 2;
        v2f aw = {}, atw = {}, at = {};
        #pragma unroll
        for (int j = 0; j < 2; ++j) {
            float qw = 0.0f, qtw = 0.0f, qt = 0.0f;
            if (valid) {
                const int idx = (row0 + j) * NC + c;
                const float p  = pred[idx];
                const float t  = target[idx];
                const float rm = rand_mat[idx];
                const float z  = (t > 0.5f) ? -p : p;   // g = sigmoid(z)
                const bool hard = (z >= LN9);           // g in [0.9, 1.0+1e-6)
                const bool drop = rm > dr;
                const float w = (hard && drop) ? 0.0f : 1.0f;
                qw = w; qtw = t * w; qt = t;
            }
            if (j == 0) { aw.x = qw; atw.x = qtw; at.x = qt; }
            else        { aw.y = qw; atw.y = qtw; at.y = qt; }
        }
        // D = A x ones + C : folds 4 batch rows into per-class accumulators
        accW  = __builtin_amdgcn_wmma_f32_16x16x4_f32(false, aw,  false, bones,
                                                      (short)0, accW,  false, false);
        accTW = __builtin_amdgcn_wmma_f32_16x16x4_f32(false, atw, false, bones,
                                                      (short)0, accTW, false, false);
        accT  = __builtin_amdgcn_wmma_f32_16x16x4_f32(false, at,  false, bones,
                                                      (short)0, accT,  false, false);
    }

    // C/D layout: VGPR r, lanes 0-15 -> M=r ; lanes 16-31 -> M=8+r. All N equal.
    if (clsLocal == 0) {   // lanes 0 and 16
        const int cbase = g * 16 + (half ? 8 : 0);
        #pragma unroll
        for (int r2 = 0; r2 < 8; ++r2) {
            const int cc = cbase + r2;
            if (cc < NC) {
                atomicAdd(&ws[cc],          accW[r2]);
                atomicAdd(&ws[NC + cc],     accTW[r2]);
                atomicAdd(&ws[2 * NC + cc], accT[r2]);
            }
        }
    }
}

__global__ void ghm_pass2(float* __restrict__ ws) {
    const int c = threadIdx.x;
    if (c >= NC) return;
    const float bc   = ws[c];           // batch_current
    const float pos  = ws[NC + c];      // pos_sum
    const float npos = ws[2 * NC + c];  // Σ target (count of t==1)
    const float bal  = 0.5f * bc;       // balance_num
    const float neg  = bc - pos;
    const float pos_gt = (pos >= bal) ? 1.0f : 0.0f;
    const float neg_gt = (neg >  bal) ? 1.0f : 0.0f;
    // target in {0,1}: cnt(target==x) = x ? npos : NB-npos
    const float cnt_maj = (pos_gt == 1.0f) ? npos : (NBf - npos);
    const float cnt_min = (neg_gt == 1.0f) ? npos : (NBf - npos);
    ws[3 * NC + c] = pos_gt;
    ws[4 * NC + c] = neg_gt;
    ws[5 * NC + c] = bal / fmaxf(cnt_maj, 1.0f);              // scale_maj
    ws[6 * NC + c] = (bc - bal) / fmaxf(cnt_min, 1.0f);       // scale_min
    ws[7 * NC + c] = (cnt_min > 0.0f) ? 1.0f : 0.0f;          // minority-present flag
}

__global__ __launch_bounds__(P3_BLOCK)
void ghm_pass3(const float* __restrict__ pred, const float* __restrict__ target,
               const float* __restrict__ rand_mat, const float* __restrict__ drate,
               const float* __restrict__ ws, float* __restrict__ out) {
    __shared__ float red[P3_BLOCK / 32];
    const int tid = blockIdx.x * P3_BLOCK + threadIdx.x;
    const int stride = P3_BLOCK * P3_GRID;     // multiple of 40 -> class invariant
    const int c = tid % NC;
    const float dr      = drate[c];
    const float pos_gt  = ws[3 * NC + c];
    const float neg_gt  = ws[4 * NC + c];
    const float smaj    = ws[5 * NC + c];
    const float smin    = ws[6 * NC + c];
    const float minflag = ws[7 * NC + c];

    float acc = 0.0f;
    const int total = NB * NC;
    for (int i = tid; i < total; i += stride) {
        const float p  = pred[i];
        const float t  = target[i];
        const float rm = rand_mat[i];
        const float z  = (t > 0.5f) ? -p : p;   // g = sigmoid(z), monotone
        const bool hard = (z >= LN9);           // g >= 0.9 (g < 1+1e-6 always)
        const bool easy = (z < -LN9);           // g < 0.1
        const bool drop = rm > dr;
        float w = (hard && drop) ? 0.0f : 1.0f;
        const bool maj  = (t == pos_gt);
        if (easy && maj) w = 0.0f;
        if (maj) w *= smaj;
        const bool mino = (t == neg_gt);
        if (mino && (minflag > 0.0f)) w *= smin;
        // stable BCE-with-logits (precise path: feeds the final loss value)
        const float bce = fmaxf(p, 0.0f) - p * t + log1pf(expf(-fabsf(p)));
        acc += bce * w;
    }

    // wave32 shuffle reduction, then cross-wave via LDS
    #pragma unroll
    for (int off = 16; off > 0; off >>= 1)
        acc += __shfl_down(acc, off, 32);
    const int lane = threadIdx.x & 31;
    const int wv   = threadIdx.x >> 5;
    if (lane == 0) red[wv] = acc;
    __syncthreads();
    if (threadIdx.x == 0) {
        float s = 0.0f;
        #pragma unroll
        for (int i = 0; i < P3_BLOCK / 32; ++i) s += red[i];
        atomicAdd(out, s * (1.0f / (NBf * (float)NC)));
    }
}

extern "C" void kernel_launch(void* const* d_in, const int* in_sizes, int n_in,
                              void* d_out, int out_size, void* d_ws, size_t ws_size,
                              hipStream_t stream) {
    const float* pred     = (const float*)d_in[0];
    const float* target   = (const float*)d_in[1];
    const float* rand_mat = (const float*)d_in[2];
    const float* drate    = (const float*)d_in[3];
    float* out = (float*)d_out;
    float* ws  = (float*)d_ws;

    ghm_init <<<1, 128, 0, stream>>>(ws, out);
    ghm_pass1<<<P1_GRID, P1_BLOCK, 0, stream>>>(pred, target, rand_mat, drate, ws);
    ghm_pass2<<<1, 64, 0, stream>>>(ws);
    ghm_pass3<<<P3_GRID, P3_BLOCK, 0, stream>>>(pred, target, rand_mat, drate, ws, out);
}